// FlattenHead_10557029613715
// MI455X (gfx1250) — compile-verified
//
#include <hip/hip_runtime.h>

// FlattenHead: compact the first seq_lens[b] rows of x[B,T,D] into d_out.
// Pure data movement (0 FLOPs) -> HBM-bandwidth bound (~11us floor at 23.3 TB/s).
// CDNA5 path: async global->LDS->global DMA (ASYNCcnt) with LDS as a bounce
// buffer; fallback is a B128 non-temporal VGPR copy.

#define B_ 16
#define T_ 4096
#define D_ 1024

#define ROWS 4            // rows (4 KB each) staged per block
#define THREADS 256       // 8 wave32 waves; one 16B slice of a row per lane

#if __has_builtin(__builtin_amdgcn_global_load_async_to_lds_b128) && \
    __has_builtin(__builtin_amdgcn_global_store_async_from_lds_b128)
#define HAVE_ASYNC_LDS 1
#else
#define HAVE_ASYNC_LDS 0
#endif

// Native clang vector types (work with nontemporal + async-LDS builtins).
typedef int   v4i __attribute__((ext_vector_type(4)));
typedef float v4f __attribute__((ext_vector_type(4)));
typedef __attribute__((address_space(1))) v4i* gv4i_p;
typedef __attribute__((address_space(3))) v4i* lv4i_p;

__device__ __forceinline__ void wait_async0() {
#if __has_builtin(__builtin_amdgcn_s_wait_asynccnt)
    __builtin_amdgcn_s_wait_asynccnt(0);
#else
    asm volatile("s_wait_asynccnt 0" ::: "memory");
#endif
}

__global__ __launch_bounds__(THREADS)
void flatten_head_compact(const float* __restrict__ x,
                          const int* __restrict__ seq_lens,
                          float* __restrict__ out) {
    const int b   = blockIdx.y;
    const int t0  = blockIdx.x * ROWS;
    const int len = seq_lens[b];          // uniform -> scalar load
    if (t0 >= len) return;

    // rows emitted before batch b (B=16: tiny uniform scalar scan)
    long long prefix = 0;
    for (int i = 0; i < b; ++i) prefix += (long long)seq_lens[i];

    const int tid   = threadIdx.x;        // 0..255, owns bytes [tid*16, tid*16+16)
    const int nrows = min(ROWS, len - t0);

#if HAVE_ASYNC_LDS
    __shared__ float stage[ROWS * D_];    // 16 KB bounce buffer, per-lane-disjoint
    auto* lds = (__attribute__((address_space(3))) char*)stage;

    // Issue all row loads (1 async B128 instruction per wave per row: 32 lanes
    // x 16B = 512B per instruction), then drain ASYNCcnt, then stream out.
    #pragma unroll
    for (int r = 0; r < ROWS; ++r) {
        if (r >= nrows) break;
        const size_t srcB = (((size_t)b * T_ + (size_t)(t0 + r)) * D_
                             + (size_t)tid * 4) * sizeof(float);
        __builtin_amdgcn_global_load_async_to_lds_b128(
            (gv4i_p)((const char*)x + srcB),
            (lv4i_p)(lds + ((size_t)r * D_ + (size_t)tid * 4) * sizeof(float)),
            0, 0);
    }
    wait_async0();                        // loads complete in order; all landed
    #pragma unroll
    for (int r = 0; r < ROWS; ++r) {
        if (r >= nrows) break;
        const size_t dstB = (((size_t)prefix + (size_t)(t0 + r)) * D_
                             + (size_t)tid * 4) * sizeof(float);
        __builtin_amdgcn_global_store_async_from_lds_b128(
            (gv4i_p)((char*)out + dstB),
            (lv4i_p)(lds + ((size_t)r * D_ + (size_t)tid * 4) * sizeof(float)),
            0, 0);
    }
    // s_endpgm performs an implicit wait-idle; LDS not reused afterwards.
#else
    // Fallback: coalesced B128 copy with non-temporal hints (read-once /
    // write-once stream should not pollute the 192MB L2).
    const v4f* __restrict__ xv = (const v4f*)x;
    v4f* __restrict__       ov = (v4f*)out;
    #pragma unroll
    for (int r = 0; r < ROWS; ++r) {
        if (r >= nrows) break;
        const size_t s = ((size_t)b * T_ + (size_t)(t0 + r)) * (D_ / 4) + tid;
        const size_t d = ((size_t)prefix + (size_t)(t0 + r)) * (D_ / 4) + tid;
        v4f v = __builtin_nontemporal_load(&xv[s]);
        __builtin_nontemporal_store(v, &ov[d]);
    }
#endif
}

extern "C" void kernel_launch(void* const* d_in, const int* in_sizes, int n_in,
                              void* d_out, int out_size, void* d_ws, size_t ws_size,
                              hipStream_t stream) {
    (void)in_sizes; (void)n_in; (void)out_size; (void)d_ws; (void)ws_size;
    const float* x   = (const float*)d_in[0];
    const int*   seq = (const int*)d_in[1];   // harness integer convention
    float*       out = (float*)d_out;

    dim3 grid(T_ / ROWS, B_);   // 1024 x 16 blocks
    dim3 block(THREADS);        // 256 threads = 8 wave32 waves
    flatten_head_compact<<<grid, block, 0, stream>>>(x, seq, out);
}